// SpecConv2D_52974126628967
// MI455X (gfx1250) — compile-verified
//
#include <hip/hip_runtime.h>

// ---------------------------------------------------------------------------
// SpecConv2D: out[b,i,j,c] = relu( sum_{r,dj in 0..2}
//                 Lambda[(3i+r)*510 + 3j+dj] * x[b, 3i+r, 3j+dj, c] )
// x: (8,510,510,64) f32, Lambda: (1, 510*510) f32, out: (8,170,170,64) f32
//
// Memory-bound: 593 MB traffic vs 252 MFLOP => ~25us floor @ 23.3 TB/s.
// Data path: double-buffered TDM tensor_load_to_lds pipeline. One block per
// output row (b,i); wave0 DMAs tile t+1 (3 x 30cols x 64ch = 23KB) while all
// 10 waves compute tile t from LDS with f32 FMAs. TENSORcnt ops are in-order
// per wave, so s_wait_tensorcnt(1) after issuing t+1 guarantees t is resident.
// ---------------------------------------------------------------------------

typedef unsigned int u32x4 __attribute__((ext_vector_type(4)));
typedef int          i32x8 __attribute__((ext_vector_type(8)));
typedef int          i32x4 __attribute__((ext_vector_type(4)));

#define B_   8
#define N_   510
#define M_   510
#define C_   64
#define O_   170                    // (510-3)/3 + 1
#define JT   10                     // output pixels per tile along j
#define NTILE (O_ / JT)             // 17 tiles per output row
#define TILE_COLS (JT * 3)          // 30 input columns per tile
#define ROW_ELEMS (TILE_COLS * C_)  // 1920 f32 per tile row (7680 B)

__global__ __launch_bounds__(JT * 32)
void spec_conv2d_tdm_kernel(const float* __restrict__ x,
                            const float* __restrict__ lam,
                            float* __restrict__ out) {
  __shared__ float smem[2][3 * ROW_ELEMS];   // ping-pong, 2 x 23040 B

  const int i    = blockIdx.x;          // output row 0..169
  const int b    = blockIdx.y;          // batch 0..7
  const int wave = threadIdx.x >> 5;    // 0..9 -> output pixel (tile*JT + wave)
  const int lane = threadIdx.x & 31;    // float2 of channels per lane

  const float* xrow = x + (((size_t)b * N_ + 3 * i) * M_) * C_;  // row 3i, col 0
  const float* lrow0 = lam + (3 * i) * M_;                       // scales, row 3i

  // Issue a TDM load of tile t (input cols 3*t*JT .. +29, 3 rows) into buf s.
  auto tdm_issue = [&](int t, int s) {
    const unsigned long long gaddr =
        (unsigned long long)(const void*)(xrow + (size_t)(3 * t * JT) * C_);
    const unsigned lds_addr = (unsigned)(unsigned long long)(const void*)smem[s];

    // D# group 0: count=1 (valid), lds_addr, global_addr[56:0], type=2
    u32x4 g0;
    g0[0] = 1u;
    g0[1] = lds_addr;
    g0[2] = (unsigned)(gaddr & 0xFFFFFFFFull);
    g0[3] = (unsigned)((gaddr >> 32) & 0x1FFFFFFull) | (2u << 30);

    // D# group 1: data_size=4B; tensor_dim0 = tile_dim0 = 1920 elems;
    //             tensor_dim1 = tile_dim1 = 3 rows; dim0_stride = 510*64 elems
    i32x8 g1;
    g1[0] = (2 << 16);                           // data_size = 4 bytes
    g1[1] = (int)((ROW_ELEMS & 0xFFFF) << 16);   // tensor_dim0[15:0]
    g1[2] = (ROW_ELEMS >> 16) | (3 << 16);       // tensor_dim0 hi | tensor_dim1
    g1[3] = (int)(ROW_ELEMS << 16);              // tensor_dim1 hi=0 | tile_dim0
    g1[4] = 3;                                   // tile_dim1=3, tile_dim2=0
    g1[5] = M_ * C_;                             // tensor_dim0_stride = 32640
    g1[6] = 0;
    g1[7] = 0;

    i32x4 gz4 = {0, 0, 0, 0};                    // 2D tensor: groups 2/3 unused
    i32x8 gz8 = {0, 0, 0, 0, 0, 0, 0, 0};
    __builtin_amdgcn_tensor_load_to_lds(g0, g1, gz4, gz4, gz8, /*cpol=*/0);
  };

  if (threadIdx.x == 0) tdm_issue(0, 0);         // prime the pipeline

  for (int t = 0; t < NTILE; ++t) {
    if (threadIdx.x == 0) {
      if (t + 1 < NTILE) {
        tdm_issue(t + 1, (t + 1) & 1);           // prefetch next tile
        __builtin_amdgcn_s_wait_tensorcnt(1);    // in-order: tile t resident
      } else {
        __builtin_amdgcn_s_wait_tensorcnt(0);    // last tile resident
      }
    }
    __syncthreads();                             // publish tile t to all waves

    const float* tile = smem[t & 1];
    const int j = t * JT + wave;                 // this wave's output pixel
    const float* lrow = lrow0 + 3 * j;           // wave-uniform 3x3 scales

    float2 acc;
    acc.x = 0.0f;
    acc.y = 0.0f;
#pragma unroll
    for (int r = 0; r < 3; ++r) {
#pragma unroll
      for (int dj = 0; dj < 3; ++dj) {
        const float s = lrow[r * M_ + dj];
        const float2 v = *(const float2*)
            (&tile[r * ROW_ELEMS + (wave * 3 + dj) * C_ + lane * 2]);
        acc.x += s * v.x;
        acc.y += s * v.y;
      }
    }
    acc.x = fmaxf(acc.x, 0.0f);
    acc.y = fmaxf(acc.y, 0.0f);

    float2* op = (float2*)(out + (((size_t)b * O_ + i) * O_ + j) * C_);
    op[lane] = acc;

    __syncthreads();   // all reads of buf t&1 done before it is refilled (t+2)
  }
}

extern "C" void kernel_launch(void* const* d_in, const int* in_sizes, int n_in,
                              void* d_out, int out_size, void* d_ws, size_t ws_size,
                              hipStream_t stream) {
  (void)in_sizes; (void)n_in; (void)d_ws; (void)ws_size; (void)out_size;
  const float* x   = (const float*)d_in[0];
  const float* lam = (const float*)d_in[1];
  float*       out = (float*)d_out;

  dim3 grid(O_, B_);            // (170, 8) = 1360 blocks, one per output row
  dim3 block(JT * 32);          // 320 threads = 10 waves, 1 wave per pixel
  spec_conv2d_tdm_kernel<<<grid, block, 0, stream>>>(x, lam, out);
}